// DeepONetCfCDecoder_50912542326892
// MI455X (gfx1250) — compile-verified
//
#include <hip/hip_runtime.h>
#include <math.h>

// ---------------------------------------------------------------------------
// DeepONet-CfC decoder for MI455X (gfx1250, wave32, WMMA).
// Dense math runs through V_WMMA_F32_16X16X4_F32 (fp32, matches fp32 ref).
// GEMM: B tile staged once per block into LDS in WMMA-fragment order =>
// inner loop is 1 pipelined global b64 (A) + 4 conflict-free ds b64 (B)
// + 4 WMMAs, 32-bit LDS addressing. ~7 GFLOP / ~9MB traffic total =>
// latency bound; weights L2-resident (192MB L2).
// ---------------------------------------------------------------------------

typedef float v2f __attribute__((ext_vector_type(2)));
typedef float v8f __attribute__((ext_vector_type(8)));

#define EPI_NONE 0
#define EPI_SILU 1
#define EPI_RES  2

// C[M,N] = A[M,K] @ B[K,N] (+bias) with optional silu / residual epilogue.
// Row-major. M % 64 == 0, N % 64 == 0, K % 4 == 0, K <= 256.
// Block = 128 threads (4 waves); block tile 64(M) x 64(N); each wave computes
// a 16x64 strip via 4 f32 WMMA accumulators.
__global__ __launch_bounds__(128) void gemm_wmma_f32(
    const float* __restrict__ A, const float* __restrict__ B,
    const float* __restrict__ bias, const float* __restrict__ resid,
    float* __restrict__ C, int M, int N, int Kd, int epi)
{
    __shared__ float blds[64 * 256];   // B tile in fragment order (64KB max)

    const int tid  = threadIdx.x;
    const int lane = tid & 31;
    const int wave = tid >> 5;
    const int half = lane >> 4;        // 0: lanes 0-15, 1: lanes 16-31
    const int l16  = lane & 15;
    const int row0 = blockIdx.y * 64 + wave * 16;
    const int col0 = blockIdx.x * 64;

    // ---- stage B tile [Kd x 64] into LDS, permuted into fragment order ----
    // Fragment addr for (kk, t, lane): ((kk*4 + t)*32 + lane)*2 + which,
    // holding pair (B[4kk+2h][col0+16t+l16], B[4kk+2h+1][col0+16t+l16]).
    {
        const int c  = tid & 63;       // column within tile (coalesced reads)
        const int t  = c >> 4;
        const int cl = c & 15;
        int kr       = tid >> 6;       // starting row: 0 or 1
        const float* P = B + (size_t)kr * N + col0 + c;
        const int nfill = Kd >> 1;     // (Kd*64)/128 iterations, stride 2 rows
        for (int i = 0; i < nfill; ++i) {
            const int kk = kr >> 2, r = kr & 3, h = r >> 1, wh = r & 1;
            blds[((((kk << 2) | t) << 5) + (h << 4) + cl) * 2 + wh] = *P;
            P += 2 * N;
            kr += 2;
        }
    }
    __syncthreads();

    v8f acc[4];
    acc[0] = v8f{}; acc[1] = v8f{}; acc[2] = v8f{}; acc[3] = v8f{};

    // A 16x4 fragment: lanes 0-15 -> row l16 K={k,k+1}; lanes 16-31 -> K={k+2,k+3}
    const float* Arow = A + (size_t)(row0 + l16) * Kd + 2 * half;
    const int nk = Kd >> 2;
    v2f a_nxt;
    a_nxt.x = Arow[0];
    a_nxt.y = Arow[1];
    for (int kk = 0; kk < nk; ++kk) {
        const v2f a = a_nxt;
        if (kk + 1 < nk) {             // software-pipeline next A fragment
            a_nxt.x = Arow[4 * (kk + 1)];
            a_nxt.y = Arow[4 * (kk + 1) + 1];
        }
        const float* bp = blds + kk * 256 + lane * 2;
        #pragma unroll
        for (int t = 0; t < 4; ++t) {
            const v2f b = *(const v2f*)(bp + t * 64);   // ds_load_b64, no conflicts
            acc[t] = __builtin_amdgcn_wmma_f32_16x16x4_f32(
                false, a, false, b, (short)0, acc[t], false, false);
        }
    }

    // C/D layout: VGPR j: lanes 0-15 -> M=j, lanes 16-31 -> M=j+8; lane = column
    #pragma unroll
    for (int j = 0; j < 8; ++j) {
        const int r = row0 + j + 8 * half;
        float* Crow = C + (size_t)r * N;
        const float* Rrow = resid ? (resid + (size_t)r * N) : nullptr;
        #pragma unroll
        for (int t = 0; t < 4; ++t) {
            const int cidx = col0 + t * 16 + l16;
            float v = acc[t][j];
            if (bias) v += bias[cidx];
            if (epi == EPI_SILU)      v = v / (1.0f + expf(-v));   // silu
            else if (epi == EPI_RES)  v += Rrow[cidx];
            Crow[cidx] = v;
        }
    }
}

// LayerNorm over rows of width 256; one wave32 per row, 8 rows per block.
__global__ __launch_bounds__(256) void layernorm256(
    const float* __restrict__ x, const float* __restrict__ g,
    const float* __restrict__ b, float* __restrict__ y, int rows)
{
    const int wave = threadIdx.x >> 5, lane = threadIdx.x & 31;
    const int row = blockIdx.x * 8 + wave;
    if (row >= rows) return;
    const float* xr = x + (size_t)row * 256;
    float v[8];
    float s = 0.f;
    #pragma unroll
    for (int i = 0; i < 8; ++i) { v[i] = xr[lane + 32 * i]; s += v[i]; }
    #pragma unroll
    for (int off = 16; off; off >>= 1) s += __shfl_xor(s, off, 32);
    const float mean = s * (1.0f / 256.0f);
    float sq = 0.f;
    #pragma unroll
    for (int i = 0; i < 8; ++i) { const float d = v[i] - mean; sq += d * d; }
    #pragma unroll
    for (int off = 16; off; off >>= 1) sq += __shfl_xor(sq, off, 32);
    const float rstd = rsqrtf(sq * (1.0f / 256.0f) + 1e-5f);
    float* yr = y + (size_t)row * 256;
    #pragma unroll
    for (int i = 0; i < 8; ++i) {
        const int e = lane + 32 * i;
        yr[e] = (v[i] - mean) * rstd * g[e] + b[e];
    }
}

// Pad trunk_in weight [60,256] -> [64,256] (zero rows 60..63).
__global__ void pad_w0(const float* __restrict__ W, float* __restrict__ Wp)
{
    const int i = blockIdx.x * blockDim.x + threadIdx.x;   // 0..16383
    const int row = i >> 8, col = i & 255;
    Wp[i] = (row < 60) ? W[row * 256 + col] : 0.f;
}

// Fused bias for collapsed projections: out[j] = b2[j] + sum_i bt[i]*W2[i,j].
__global__ void fuse_bias(const float* __restrict__ bt, const float* __restrict__ W2,
                          const float* __restrict__ b2, float* __restrict__ out)
{
    const int j = blockIdx.x * blockDim.x + threadIdx.x;   // 0..255
    float s = b2[j];
    for (int i = 0; i < 256; ++i) s += bt[i] * W2[i * 256 + j];   // coalesced across j
    out[j] = s;
}

// Per-query featurization: searchsorted + fourier/time/comp features (60 + 4 pad).
// NOTE: the reference's rel-pos bias collapses to a constant (LayerNorm over a
// size-1 axis == its bias), and softmax is shift-invariant, so it is dropped.
__global__ void build_query_feat(
    const float* __restrict__ xy, const float* __restrict__ tq,
    const int* __restrict__ c, const float* __restrict__ sensor_time,
    const float* __restrict__ Wt, const float* __restrict__ bt,
    const float* __restrict__ emb, float* __restrict__ X0,
    int* __restrict__ idx_out, int Nq)
{
    const int n = blockIdx.x * blockDim.x + threadIdx.x;
    if (n >= Nq) return;
    const float t = tq[n];
    // searchsorted(side='right') - 1, clipped to [0,127]
    int lo = 0, hi = 128;
    while (lo < hi) { const int mid = (lo + hi) >> 1; if (sensor_time[mid] <= t) lo = mid + 1; else hi = mid; }
    int idx = lo - 1; if (idx < 0) idx = 0; if (idx > 127) idx = 127;
    idx_out[n] = idx;
    float dt = t - sensor_time[idx]; if (dt < 0.f) dt = 0.f;

    const float x = xy[2 * n], y = xy[2 * n + 1];
    float* f = X0 + (size_t)n * 64;
    const float TWO_PI = 6.28318530717958647692f;
    #pragma unroll
    for (int k = 0; k < 8; ++k) {                // L = 1.0
        const float w = TWO_PI * (float)(k + 1);
        float sx, cx, sy, cy;
        sincosf(w * x, &sx, &cx);
        sincosf(w * y, &sy, &cy);
        f[4 * k + 0] = sx; f[4 * k + 1] = cx; f[4 * k + 2] = sy; f[4 * k + 3] = cy;
    }
    #pragma unroll
    for (int m = 0; m < 2; ++m) {                // T_TOTAL = 5.0
        const float w = (TWO_PI / 5.0f) * (float)(m + 1);
        float sa, ca;
        sincosf(w * t, &sa, &ca);
        f[32 + 2 * m] = sa; f[33 + 2 * m] = ca;
    }
    #pragma unroll
    for (int d = 0; d < 16; ++d) f[36 + d] = dt * Wt[d] + bt[d];   // time_proj [1,16]
    const int cc = c[n];
    #pragma unroll
    for (int j = 0; j < 8; ++j) f[52 + j] = emb[cc * 8 + j];
    f[60] = f[61] = f[62] = f[63] = 0.f;
}

// Cross-attention: one 64-thread block (2 waves) per query; K=64 tokens, H=256.
__global__ __launch_bounds__(64) void attention(
    const float* __restrict__ qb, const float* __restrict__ kall,
    const float* __restrict__ vall, const int* __restrict__ idx,
    float* __restrict__ ctx, int Nq)
{
    __shared__ float qs[256];
    __shared__ float sc[64];
    const int n = blockIdx.x;
    const int t = threadIdx.x;       // 0..63
    const float* qr = qb + (size_t)n * 256;
    qs[t] = qr[t]; qs[t + 64] = qr[t + 64]; qs[t + 128] = qr[t + 128]; qs[t + 192] = qr[t + 192];
    __syncthreads();
    const int id = idx[n];
    const float* kr = kall + ((size_t)id * 64 + t) * 256;
    float s = 0.f;
    #pragma unroll 8
    for (int d = 0; d < 256; ++d) s += qs[d] * kr[d];
    s *= 0.0625f;                    // 1/sqrt(H), H=256
    sc[t] = s;
    __syncthreads();
    float m = -3.0e38f;
    for (int j = 0; j < 64; ++j) m = fmaxf(m, sc[j]);
    const float e = expf(s - m);
    __syncthreads();                 // everyone done reading scores
    sc[t] = e;
    __syncthreads();
    float sum = 0.f;
    for (int j = 0; j < 64; ++j) sum += sc[j];
    const float inv = 1.0f / sum;
    const float* vbase = vall + (size_t)id * 64 * 256;
    float* cr = ctx + (size_t)n * 256;
    #pragma unroll
    for (int i = 0; i < 4; ++i) {
        const int d = t + 64 * i;    // coalesced across threads
        float acc = 0.f;
        for (int j = 0; j < 64; ++j) acc += sc[j] * vbase[j * 256 + d];
        cr[d] = acc * inv;
    }
}

// Final low-rank bilinear fuse for selected component.
__global__ void final_fuse(
    const float* __restrict__ tb, const float* __restrict__ bb,
    const int* __restrict__ c, const float* __restrict__ log_temp,
    const float* __restrict__ cs, const float* __restrict__ cbias,
    float* __restrict__ out, int Nq)
{
    const int n = blockIdx.x * blockDim.x + threadIdx.x;
    if (n >= Nq) return;
    const int cc = c[n];
    const float* tr = tb + (size_t)n * 192 + cc * 64;
    const float* br = bb + (size_t)n * 192 + cc * 64;
    float s = 0.f;
    #pragma unroll 8
    for (int i = 0; i < 64; ++i) s += tr[i] * br[i];
    out[n] = s * expf(log_temp[0]) * cs[cc] + cbias[cc];
}

// ------------------------------- host side ---------------------------------

static inline void launch_gemm(const float* A, const float* B, const float* bias,
                               const float* resid, float* C, int M, int N, int K,
                               int epi, hipStream_t s)
{
    dim3 grid(N / 64, M / 64);
    gemm_wmma_f32<<<grid, dim3(128), 0, s>>>(A, B, bias, resid, C, M, N, K, epi);
}

extern "C" void kernel_launch(void* const* d_in, const int* in_sizes, int n_in,
                              void* d_out, int out_size, void* d_ws, size_t ws_size,
                              hipStream_t stream)
{
    // Input order: setup_inputs() dict order, params flattened in insertion order.
    const float* xy          = (const float*)d_in[0];   // [4096,2]
    const float* t_q         = (const float*)d_in[1];   // [4096]
    const int*   comp        = (const int*)  d_in[2];   // [4096]
    const float* h_states    = (const float*)d_in[3];   // [128,64,256]
    const float* sensor_time = (const float*)d_in[4];   // [128]
    // d_in[5] = sensor_pos (unused: rel-pos bias is softmax-invariant constant)
    const float* timeW   = (const float*)d_in[6];   // [1,16]
    const float* timeB   = (const float*)d_in[7];   // [16]
    const float* compEmb = (const float*)d_in[8];   // [3,8]
    const float* W0      = (const float*)d_in[9];   // trunk_in.W [60,256]
    const float* b0      = (const float*)d_in[10];  // trunk_in.b
    const float* blk_lng[2]  = {(const float*)d_in[11], (const float*)d_in[17]};
    const float* blk_lnb[2]  = {(const float*)d_in[12], (const float*)d_in[18]};
    const float* blk_f1W[2]  = {(const float*)d_in[13], (const float*)d_in[19]};
    const float* blk_f1b[2]  = {(const float*)d_in[14], (const float*)d_in[20]};
    const float* blk_f2W[2]  = {(const float*)d_in[15], (const float*)d_in[21]};
    const float* blk_f2b[2]  = {(const float*)d_in[16], (const float*)d_in[22]};
    const float* bnormG  = (const float*)d_in[23];
    const float* bnormB  = (const float*)d_in[24];
    const float* tokW    = (const float*)d_in[25];
    const float* tokB    = (const float*)d_in[26];
    const float* qW      = (const float*)d_in[27];
    const float* qB      = (const float*)d_in[28];
    const float* kW      = (const float*)d_in[29];
    const float* kB      = (const float*)d_in[30];
    const float* vW      = (const float*)d_in[31];
    const float* vB      = (const float*)d_in[32];
    // d_in[33..38] = relpos params (unused, see note above)
    const float* ctxLnG  = (const float*)d_in[39];
    const float* ctxLnB  = (const float*)d_in[40];
    const float* ctxF1W  = (const float*)d_in[41];
    const float* ctxF1b  = (const float*)d_in[42];
    const float* ctxF2W  = (const float*)d_in[43];
    const float* ctxF2b  = (const float*)d_in[44];
    const float* toutW   = (const float*)d_in[45];  // [256,192]
    const float* toutB   = (const float*)d_in[46];
    const float* bprojW  = (const float*)d_in[47];  // [256,192]
    const float* bprojB  = (const float*)d_in[48];
    const float* logTemp = (const float*)d_in[49];
    const float* compSc  = (const float*)d_in[50];
    const float* compBi  = (const float*)d_in[51];

    const int N  = 4096;
    const int SK = 128 * 64;   // 8192 branch token rows

    // Workspace layout (float offsets); ~46 MB total.
    float* ws = (float*)d_ws;
    float* kall  = ws + 0;          // 2097152
    float* vall  = ws + 2097152;    // 2097152
    float* Wfk   = ws + 4194304;    // 65536
    float* Wfv   = ws + 4259840;    // 65536
    float* bfk   = ws + 4325376;    // 256
    float* bfv   = ws + 4325632;    // 256
    float* X0    = ws + 4325888;    // 262144
    float* W0pad = ws + 4588032;    // 16384
    float* Tf    = ws + 4604416;    // 1048576 (trunk_feat)
    float* Ybuf  = ws + 5652992;    // 1048576
    float* Zbuf  = ws + 6701568;    // 1048576
    float* qbuf  = ws + 7750144;    // 1048576
    float* tbas  = ws + 8798720;    // 786432 (trunk_basis [4096,192])
    float* ctxb  = ws + 9585152;    // 1048576
    float* bbas  = ws + 10633728;   // 786432 (branch_basis)
    int*   idxb  = (int*)(ws + 11420160);  // 4096 ints
    (void)ws_size; (void)n_in; (void)in_sizes; (void)out_size;

    // --- prep ---
    pad_w0<<<64, 256, 0, stream>>>(W0, W0pad);
    build_query_feat<<<N / 256, 256, 0, stream>>>(xy, t_q, comp, sensor_time,
                                                  timeW, timeB, compEmb, X0, idxb, N);

    // --- branch precompute: fold tok_proj into k/v projections ---
    launch_gemm(tokW, kW, nullptr, nullptr, Wfk, 256, 256, 256, EPI_NONE, stream);
    launch_gemm(tokW, vW, nullptr, nullptr, Wfv, 256, 256, 256, EPI_NONE, stream);
    fuse_bias<<<1, 256, 0, stream>>>(tokB, kW, kB, bfk);
    fuse_bias<<<1, 256, 0, stream>>>(tokB, vW, vB, bfv);
    launch_gemm(h_states, Wfk, bfk, nullptr, kall, SK, 256, 256, EPI_NONE, stream);
    launch_gemm(h_states, Wfv, bfv, nullptr, vall, SK, 256, 256, EPI_NONE, stream);

    // --- trunk MLP ---
    launch_gemm(X0, W0pad, b0, nullptr, Tf, N, 256, 64, EPI_SILU, stream);
    for (int b = 0; b < 2; ++b) {
        layernorm256<<<N / 8, 256, 0, stream>>>(Tf, blk_lng[b], blk_lnb[b], Ybuf, N);
        launch_gemm(Ybuf, blk_f1W[b], blk_f1b[b], nullptr, Zbuf, N, 256, 256, EPI_SILU, stream);
        launch_gemm(Zbuf, blk_f2W[b], blk_f2b[b], Tf, Tf, N, 256, 256, EPI_RES, stream);
    }
    launch_gemm(Tf, toutW, toutB, nullptr, tbas, N, 192, 256, EPI_NONE, stream);

    // --- query projection + attention ---
    layernorm256<<<N / 8, 256, 0, stream>>>(Tf, bnormG, bnormB, Ybuf, N);
    launch_gemm(Ybuf, qW, qB, nullptr, qbuf, N, 256, 256, EPI_NONE, stream);
    attention<<<N, 64, 0, stream>>>(qbuf, kall, vall, idxb, ctxb, N);

    // --- context residual MLP + branch head ---
    layernorm256<<<N / 8, 256, 0, stream>>>(ctxb, ctxLnG, ctxLnB, Ybuf, N);
    launch_gemm(Ybuf, ctxF1W, ctxF1b, nullptr, Zbuf, N, 256, 256, EPI_SILU, stream);
    launch_gemm(Zbuf, ctxF2W, ctxF2b, ctxb, ctxb, N, 256, 256, EPI_RES, stream);
    launch_gemm(ctxb, bprojW, bprojB, nullptr, bbas, N, 192, 256, EPI_NONE, stream);

    // --- final bilinear fuse ---
    final_fuse<<<N / 256, 256, 0, stream>>>(tbas, bbas, comp, logTemp, compSc, compBi,
                                            (float*)d_out, N);
}